// MultiHeadAttention_5394478924592
// MI455X (gfx1250) — compile-verified
//
#include <hip/hip_runtime.h>

// MHA: x[8,1024,768] -> (out[8,1024,768], attn[8,12,1024,1024]), fp32.
// 3 kernels: qkv GEMM, fused scores+softmax+attn@V (attn strip LDS-resident,
// written to HBM exactly once), proj GEMM. All matmuls via
// V_WMMA_F32_16X16X4_F32; staging via async global->LDS copies (ASYNCcnt).

#define BM 64
#define BN 64
#define BK 32
#define AP 65  // GEMM As pitch (transposed scatter, conflict-free)
#define GBP 68 // GEMM Bs pitch (row copies, 16B aligned)

#define SP 33  // fused: Ss[j][m] pitch (32 rows + 1)
#define QP 33  // fused: Qs[d][m] pitch
#define KP 65  // fused: Ks[d][j] pitch (scores)
#define VP 68  // fused: Vs[j][d] pitch (attn@V)

typedef float v2f __attribute__((ext_vector_type(2)));
typedef float v8f __attribute__((ext_vector_type(8)));

#if defined(__has_builtin)
#if __has_builtin(__builtin_amdgcn_global_load_async_to_lds_b32) && \
    __has_builtin(__builtin_amdgcn_global_load_async_to_lds_b128)
#define USE_ASYNC 1
#endif
#endif

#ifdef USE_ASYNC
typedef int v4i_ __attribute__((vector_size(16)));
typedef __attribute__((address_space(1))) int* g_i32p;
typedef __attribute__((address_space(3))) int* l_i32p;
typedef __attribute__((address_space(1))) v4i_* g_v4ip;
typedef __attribute__((address_space(3))) v4i_* l_v4ip;

__device__ __forceinline__ void async_cp_b32(const float* g, float* l) {
  __builtin_amdgcn_global_load_async_to_lds_b32((g_i32p)const_cast<float*>(g),
                                                (l_i32p)l, 0, 0);
}
__device__ __forceinline__ void async_cp_b128(const float* g, float* l) {
  __builtin_amdgcn_global_load_async_to_lds_b128((g_v4ip)const_cast<float*>(g),
                                                 (l_v4ip)l, 0, 0);
}
__device__ __forceinline__ void wait_async_lds() {
#if __has_builtin(__builtin_amdgcn_s_wait_asynccnt)
  __builtin_amdgcn_s_wait_asynccnt(0);
#else
  asm volatile("s_wait_asynccnt 0" ::: "memory");
#endif
}
#endif

// Copy a float4 worth of A-row into LDS transposed (scatter 4 scalars).
__device__ __forceinline__ void stage_t4(const float* gp, float* lds, int pitch,
                                         int krow, int col) {
#ifdef USE_ASYNC
#pragma unroll
  for (int c = 0; c < 4; ++c) async_cp_b32(gp + c, &lds[(krow + c) * pitch + col]);
#else
  const float4 v = *(const float4*)gp;
  lds[(krow + 0) * pitch + col] = v.x;
  lds[(krow + 1) * pitch + col] = v.y;
  lds[(krow + 2) * pitch + col] = v.z;
  lds[(krow + 3) * pitch + col] = v.w;
#endif
}
__device__ __forceinline__ void stage_row4(const float* gp, float* ldst) {
#ifdef USE_ASYNC
  async_cp_b128(gp, ldst);
#else
  *(float4*)ldst = *(const float4*)gp;
#endif
}
__device__ __forceinline__ void stage_fence() {
#ifdef USE_ASYNC
  wait_async_lds();
#endif
  __syncthreads();
}

// ---------------- plain GEMM: C = alpha*A@B (+bias) ----------------
__global__ __launch_bounds__(256) void gemm_wmma(
    const float* __restrict__ A, const float* __restrict__ Bp,
    float* __restrict__ Cp, const float* __restrict__ bias,
    int K, int lda, int ldb, int ldc, float alpha) {
  __shared__ float As[BK * AP];
  __shared__ float Bs[BK * GBP];

  const int t = threadIdx.x;
  const int lane = t & 31;
  const int w = t >> 5;
  const int wm = (w & 3) * 16;
  const int wn = (w >> 2) * 32;
  const int n0 = blockIdx.x * BN;
  const int m0 = blockIdx.y * BM;

  v8f acc0 = {}, acc1 = {};
  const int nk = K / BK;
  for (int ks = 0; ks < nk; ++ks) {
    const int k0 = ks * BK;
    {  // A tile transposed: As[k][m]
      const int kq = t & 7, i = t >> 3;
#pragma unroll
      for (int r = 0; r < BM; r += 32)
        stage_t4(&A[(size_t)(m0 + r + i) * lda + k0 + kq * 4], As, AP, kq * 4,
                 r + i);
    }
    {  // B tile rows: Bs[k][n]
      const int nq = t & 15, k = t >> 4;
#pragma unroll
      for (int r = 0; r < BK; r += 16)
        stage_row4(&Bp[(size_t)(k0 + k + r) * ldb + n0 + nq * 4],
                   &Bs[(k + r) * GBP + nq * 4]);
    }
    stage_fence();

    const int lm = wm + (lane & 15);
    const int ln = wn + (lane & 15);
    const int kh = (lane >> 4) << 1;
#pragma unroll
    for (int kk = 0; kk < BK; kk += 4) {
      v2f a, b0, b1;
      a.x = As[(kk + kh) * AP + lm];
      a.y = As[(kk + kh + 1) * AP + lm];
      b0.x = Bs[(kk + kh) * GBP + ln];
      b0.y = Bs[(kk + kh + 1) * GBP + ln];
      b1.x = Bs[(kk + kh) * GBP + ln + 16];
      b1.y = Bs[(kk + kh + 1) * GBP + ln + 16];
      acc0 = __builtin_amdgcn_wmma_f32_16x16x4_f32(false, a, false, b0,
                                                   (short)0, acc0, false, false);
      acc1 = __builtin_amdgcn_wmma_f32_16x16x4_f32(false, a, false, b1,
                                                   (short)0, acc1, false, false);
    }
    __syncthreads();
  }

  const int cn0 = n0 + wn + (lane & 15);
  const int mb = m0 + wm + ((lane >> 4) << 3);
  const float bv0 = bias ? bias[cn0] : 0.0f;
  const float bv1 = bias ? bias[cn0 + 16] : 0.0f;
#pragma unroll
  for (int r = 0; r < 8; ++r) {
    Cp[(size_t)(mb + r) * ldc + cn0] = acc0[r] * alpha + bv0;
    Cp[(size_t)(mb + r) * ldc + cn0 + 16] = acc1[r] * alpha + bv1;
  }
}

// ------------- fused scores + softmax + attn@V -------------
// Block: 256 threads, one (b,h), 32 query rows. Attn strip lives in LDS,
// written to global exactly once.
__global__ __launch_bounds__(256) void attn_fused(
    const float* __restrict__ qkv, float* __restrict__ attnO,
    float* __restrict__ oh) {
  __shared__ float Qs[64 * QP];     // Qs[d][m]  (64 x 32)
  __shared__ float KVs[64 * VP];    // Ks[d][j] (pitch KP) / Vs[j][d] (pitch VP)
  __shared__ float Ss[1024 * SP];   // attn strip [j][m]
  __shared__ float rowred[8 * 32];
  __shared__ float rowval[32];

  const int t = threadIdx.x, lane = t & 31, w = t >> 5;
  const int l15 = lane & 15;
  const int kh = (lane >> 4) << 1;

  const int z = blockIdx.y, b = z / 12, h = z % 12;
  const int ms = blockIdx.x * 32;
  const float* Qb = qkv + (size_t)b * 1024 * 2304 + h * 64 + (size_t)ms * 2304;
  const float* Kb = qkv + (size_t)b * 1024 * 2304 + h * 64 + 768;
  const float* Vb = Kb + 768;
  float* aG = attnO + (size_t)z * 1024 * 1024 + (size_t)ms * 1024;
  float* oG = oh + ((size_t)b * 1024 + ms) * 768 + h * 64;

  // ---- stage Q strip transposed: Qs[d][m]
  {
    const int kq = t & 15, i0 = t >> 4;
#pragma unroll
    for (int r = 0; r < 32; r += 16)
      stage_t4(&Qb[(size_t)(r + i0) * 2304 + kq * 4], Qs, QP, kq * 4, r + i0);
  }

  // ---- phase 1: logits into Ss (16 tiles of 64 cols)
  const int wmS = (w & 1) * 16;   // m sub-tile
  const int jnS = (w >> 1) * 16;  // j sub-tile within 64
  for (int jt = 0; jt < 16; ++jt) {
    {  // stage K tile transposed: Ks[d][j]
      const int kq = t & 15, j0 = t >> 4;
#pragma unroll
      for (int r = 0; r < 64; r += 16)
        stage_t4(&Kb[(size_t)(jt * 64 + r + j0) * 2304 + kq * 4], KVs, KP,
                 kq * 4, r + j0);
    }
    stage_fence();

    v8f acc = {};
#pragma unroll
    for (int kk = 0; kk < 64; kk += 4) {
      v2f a, bf;
      a.x = Qs[(kk + kh) * QP + wmS + l15];
      a.y = Qs[(kk + kh + 1) * QP + wmS + l15];
      bf.x = KVs[(kk + kh) * KP + jnS + l15];
      bf.y = KVs[(kk + kh + 1) * KP + jnS + l15];
      acc = __builtin_amdgcn_wmma_f32_16x16x4_f32(false, a, false, bf,
                                                  (short)0, acc, false, false);
    }
    const int jg = jt * 64 + jnS + l15;
    const int mB = wmS + ((lane >> 4) << 3);
#pragma unroll
    for (int r = 0; r < 8; ++r) Ss[(size_t)jg * SP + mB + r] = acc[r] * 0.125f;
    __syncthreads();
  }

  // ---- phase 2: softmax over Ss columns (each m-row of attn)
  {
    const int m = t & 31, jc = t >> 5;
    float mx = -3.4e38f;
    for (int j = jc * 128; j < jc * 128 + 128; ++j)
      mx = fmaxf(mx, Ss[(size_t)j * SP + m]);
    rowred[jc * 32 + m] = mx;
    __syncthreads();
    if (t < 32) {
      float v = rowred[t];
      for (int c = 1; c < 8; ++c) v = fmaxf(v, rowred[c * 32 + t]);
      rowval[t] = v;
    }
    __syncthreads();
    const float rm = rowval[m];
    float s = 0.f;
    for (int j = jc * 128; j < jc * 128 + 128; ++j) {
      const float e = __expf(Ss[(size_t)j * SP + m] - rm);
      Ss[(size_t)j * SP + m] = e;
      s += e;
    }
    rowred[jc * 32 + m] = s;
    __syncthreads();
    if (t < 32) {
      float v = 0.f;
      for (int c = 0; c < 8; ++c) v += rowred[c * 32 + t];
      rowval[t] = 1.0f / v;
    }
    __syncthreads();
    const float inv = rowval[m];
    for (int j = jc * 128; j < jc * 128 + 128; ++j)
      Ss[(size_t)j * SP + m] *= inv;
  }
  __syncthreads();

  // ---- emit attn strip to global (coalesced float4, single HBM write)
  for (int p = 0; p < 32; ++p) {
    const int f = p * 256 + t;       // float4 index in 32x1024 strip
    const int m = f >> 8;            // 256 float4 per row
    const int c4 = (f & 255) * 4;
    float4 vv;
    vv.x = Ss[(size_t)(c4 + 0) * SP + m];
    vv.y = Ss[(size_t)(c4 + 1) * SP + m];
    vv.z = Ss[(size_t)(c4 + 2) * SP + m];
    vv.w = Ss[(size_t)(c4 + 3) * SP + m];
    *(float4*)&aG[(size_t)m * 1024 + c4] = vv;
  }

  // ---- phase 3: out_head = attn_strip @ V (A-frags straight from Ss)
  const int mtA = (w & 1) * 16;
  const int dtA = (w >> 1) * 16;
  v8f acc = {};
  for (int jt = 0; jt < 16; ++jt) {
    {  // stage V tile rows: Vs[j][d]
      const int dq = t & 15, j0 = t >> 4;
#pragma unroll
      for (int r = 0; r < 64; r += 16)
        stage_row4(&Vb[(size_t)(jt * 64 + r + j0) * 2304 + dq * 4],
                   &KVs[(r + j0) * VP + dq * 4]);
    }
    stage_fence();
#pragma unroll
    for (int kk = 0; kk < 64; kk += 4) {
      v2f a, bf;
      const int jr = jt * 64 + kk + kh;
      a.x = Ss[(size_t)jr * SP + mtA + l15];
      a.y = Ss[(size_t)(jr + 1) * SP + mtA + l15];
      bf.x = KVs[(kk + kh) * VP + dtA + l15];
      bf.y = KVs[(kk + kh + 1) * VP + dtA + l15];
      acc = __builtin_amdgcn_wmma_f32_16x16x4_f32(false, a, false, bf,
                                                  (short)0, acc, false, false);
    }
    __syncthreads();
  }
  const int cd = dtA + l15;
  const int mB = mtA + ((lane >> 4) << 3);
#pragma unroll
  for (int r = 0; r < 8; ++r) oG[(size_t)(mB + r) * 768 + cd] = acc[r];
}

extern "C" void kernel_launch(void* const* d_in, const int* in_sizes, int n_in,
                              void* d_out, int out_size, void* d_ws, size_t ws_size,
                              hipStream_t stream) {
  const float* x      = (const float*)d_in[0];  // [8,1024,768]
  const float* w_qkv  = (const float*)d_in[1];  // [768,2304]
  const float* w_proj = (const float*)d_in[2];  // [768,768]
  const float* b_proj = (const float*)d_in[3];  // [768]

  float* out  = (float*)d_out;                  // [8,1024,768]
  float* attn = out + (size_t)8 * 1024 * 768;   // [8,12,1024,1024]

  float* qkv = (float*)d_ws;                    // [8192,2304]
  float* oh  = qkv + (size_t)8192 * 2304;       // [8192,768] head outputs

  const dim3 blk(256);

  // 1) qkv = x @ w_qkv            M=8192 N=2304 K=768
  gemm_wmma<<<dim3(2304 / BN, 8192 / BM, 1), blk, 0, stream>>>(
      x, w_qkv, qkv, nullptr, 768, 768, 2304, 2304, 1.0f);

  // 2) fused scores+softmax+attn@V: 32-row strips x 96 heads
  attn_fused<<<dim3(32, 96), blk, 0, stream>>>(qkv, attn, oh);

  // 3) out = out_head @ w_proj + b_proj   M=8192 N=768 K=768
  gemm_wmma<<<dim3(768 / BN, 8192 / BM, 1), blk, 0, stream>>>(
      oh, w_proj, out, b_proj, 768, 768, 768, 768, 1.0f);
}